// MeanAggregator_9182640078905
// MI455X (gfx1250) — compile-verified
//
#include <hip/hip_runtime.h>
#include <stdint.h>

// MeanAggregator: out[b,:] = (1/|unique(nbrs[b])|) * sum_{u in unique} features[u,:]
// B=8192, K=10, U=32768, D=128 (fp32). Gather-and-mean, done with the TDM
// (tensor_load_to_lds, gather mode) — CDNA5's embedding-gather DMA path.
constexpr int B_TGT   = 8192;
constexpr int K_NBR   = 10;
constexpr int U_NODES = 32768;
constexpr int D_FEAT  = 128;

constexpr int WAVES_PER_BLOCK = 4;
constexpr int BLOCK = 32 * WAVES_PER_BLOCK;          // 128 threads = 4 wave32
constexpr int ROW_BYTES = D_FEAT * 4;                // 512 B per feature row
constexpr int LDS_PER_WAVE = K_NBR * ROW_BYTES;      // 5 KB staging per wave

typedef __attribute__((ext_vector_type(4))) unsigned int u32x4;
typedef __attribute__((ext_vector_type(8))) int          i32x8;
typedef __attribute__((ext_vector_type(4))) int          i32x4;

__device__ __forceinline__ unsigned lds_off_u32(const void* p) {
    // Generic pointers to LDS carry the wave-relative LDS byte offset in
    // addr[31:0] (ISA 10.2 aperture mapping).
    return (unsigned)(uintptr_t)p;
}

__device__ __forceinline__ void compiler_mem_barrier() {
    asm volatile("" ::: "memory");
}

__device__ __forceinline__ void wait_tensor_zero() {
#if __has_builtin(__builtin_amdgcn_s_wait_tensorcnt)
    __builtin_amdgcn_s_wait_tensorcnt(0);
#else
    asm volatile("s_wait_tensorcnt 0x0" ::: "memory");
#endif
}

__device__ __forceinline__ void wait_async_zero() {
#if __has_builtin(__builtin_amdgcn_s_wait_asynccnt)
    __builtin_amdgcn_s_wait_asynccnt(0);
#else
    asm volatile("s_wait_asynccnt 0x0" ::: "memory");
#endif
}

__global__ __launch_bounds__(BLOCK) void mean_agg_tdm_kernel(
    const int* __restrict__ neighbors,   // [B,K] int32
    const float* __restrict__ features,  // [U,D] fp32
    float* __restrict__ out)             // [B,D] fp32
{
    __shared__ __align__(16) char smem[WAVES_PER_BLOCK * LDS_PER_WAVE];

    const int lane = threadIdx.x & 31;
    const int wave = threadIdx.x >> 5;
    const int b    = blockIdx.x * WAVES_PER_BLOCK + wave;   // grid sized exactly
    const int bu   = __builtin_amdgcn_readfirstlane(b);

    // ---- neighbor ids, forced into SGPRs: dedupe + descriptor build = SALU ----
    const int* nb = neighbors + (long long)bu * K_NBR;
    int idx[K_NBR];
#pragma unroll
    for (int j = 0; j < K_NBR; ++j)
        idx[j] = __builtin_amdgcn_readfirstlane(nb[j]);

    // ---- dup flags: set semantics of .at[rows, nbrs].set(1.0) ----
    int dup[K_NBR];
    int n = K_NBR;
#pragma unroll
    for (int j = 0; j < K_NBR; ++j) {
        int d = 0;
#pragma unroll
        for (int i = 0; i < K_NBR; ++i)
            if (i < j) d |= (idx[i] == idx[j]) ? 1 : 0;
        dup[j] = d;
        n -= d;
    }

    char* wbase = smem + wave * LDS_PER_WAVE;
    const unsigned ldsb =
        (unsigned)__builtin_amdgcn_readfirstlane((int)lds_off_u32(wbase));

#if __has_builtin(__builtin_amdgcn_tensor_load_to_lds)
    // ---- one TDM gather per wave: 10 rows x 512B -> LDS (dupes included,
    //      zero-weighted in the reduction) ----
    {
        const unsigned long long ga = (unsigned long long)(uintptr_t)features;

        u32x4 g0;
        g0.x = 0x80000001u;                       // count=1 | gather_mode=1 | 16-bit indices
        g0.y = ldsb;                              // lds_addr (bytes)
        g0.z = (unsigned)ga;                      // global_addr[31:0]
        g0.w = (unsigned)(ga >> 32) | (2u << 30); // global_addr[56:32] | type=2

        i32x8 g1;
        g1[0] = 0x00020000;                              // data_size=2 -> 4B elems
        g1[1] = (int)((unsigned)D_FEAT << 16);           // tensor_dim0 = 128
        g1[2] = (int)((unsigned)(U_NODES & 0xFFFF) << 16);         // tensor_dim1 lo16
        g1[3] = ((U_NODES >> 16) & 0xFFFF) | (int)((unsigned)D_FEAT << 16); // dim1 hi | tile_dim0=128
        g1[4] = K_NBR;                                   // tile_dim1 = #gather indices
        g1[5] = D_FEAT;                                  // tensor_dim0_stride = 128
        g1[6] = 0;
        g1[7] = 0;

        i32x4 g2, g3;                                    // 16-bit row indices, 2/dword
        g2[0] = (idx[0] & 0xFFFF) | (idx[1] << 16);
        g2[1] = (idx[2] & 0xFFFF) | (idx[3] << 16);
        g2[2] = (idx[4] & 0xFFFF) | (idx[5] << 16);
        g2[3] = (idx[6] & 0xFFFF) | (idx[7] << 16);
        g3[0] = (idx[8] & 0xFFFF) | (idx[9] << 16);
        g3[1] = 0; g3[2] = 0; g3[3] = 0;

#if __clang_major__ >= 23
        i32x8 g4 = {0, 0, 0, 0, 0, 0, 0, 0};
        __builtin_amdgcn_tensor_load_to_lds(g0, g1, g2, g3, g4, 0);
#else
        __builtin_amdgcn_tensor_load_to_lds(g0, g1, g2, g3, 0);
#endif
        wait_tensor_zero();
    }
#else
    // ---- fallback: 10 async global->LDS row copies (proven to assemble) ----
    {
        const unsigned lane_lds = ldsb + (unsigned)(lane * 16);
#pragma unroll
        for (int j = 0; j < K_NBR; ++j) {
            unsigned long long gaddr = (unsigned long long)(uintptr_t)
                (features + (long long)idx[j] * D_FEAT + lane * 4);
            unsigned ldst = lane_lds + (unsigned)(j * ROW_BYTES);
            asm volatile("global_load_async_to_lds_b128 %0, %1, off"
                         :: "v"(ldst), "v"(gaddr) : "memory");
        }
        wait_async_zero();
    }
#endif

    compiler_mem_barrier();   // force LDS reads to happen after the DMA wait

    // ---- weighted reduce: lane owns columns [4*lane, 4*lane+4) ----
    const float4* lsrc = reinterpret_cast<const float4*>(wbase) + lane;
    float4 acc = make_float4(0.f, 0.f, 0.f, 0.f);
#pragma unroll
    for (int j = 0; j < K_NBR; ++j) {
        const float w = dup[j] ? 0.0f : 1.0f;     // dupes contribute nothing
        float4 f = lsrc[j * (ROW_BYTES / 16)];
        acc.x = fmaf(w, f.x, acc.x);
        acc.y = fmaf(w, f.y, acc.y);
        acc.z = fmaf(w, f.z, acc.z);
        acc.w = fmaf(w, f.w, acc.w);
    }

    const float s = 1.0f / (float)n;
    float4 r = make_float4(acc.x * s, acc.y * s, acc.z * s, acc.w * s);
    *reinterpret_cast<float4*>(out + (long long)b * D_FEAT + lane * 4) = r;
}

extern "C" void kernel_launch(void* const* d_in, const int* in_sizes, int n_in,
                              void* d_out, int out_size, void* d_ws, size_t ws_size,
                              hipStream_t stream) {
    const int*   neighbors = (const int*)d_in[0];     // [B,K]
    const float* features  = (const float*)d_in[1];   // [U,D]
    float*       out       = (float*)d_out;           // [B,D]
    (void)in_sizes; (void)n_in; (void)out_size; (void)d_ws; (void)ws_size;

    const int grid = B_TGT / WAVES_PER_BLOCK;         // 2048 blocks x 128 threads
    hipLaunchKernelGGL(mean_agg_tdm_kernel, dim3(grid), dim3(BLOCK), 0, stream,
                       neighbors, features, out);
}